// PointNet2Module_63599875719176
// MI455X (gfx1250) — compile-verified
//
#include <hip/hip_runtime.h>
#include <hip/hip_bf16.h>

typedef __attribute__((ext_vector_type(2))) float v2f;
typedef __attribute__((ext_vector_type(8))) float v8f;

#define BN_EPS 1e-5f

// ---------------------------------------------------------------------------
// Farthest point sampling. One block per batch. Matches reference: start at
// index 0, record 'far' at top of iteration, argmax picks first (lowest idx)
// max. dist kept in LDS (N <= 4096 -> 16KB).
// ---------------------------------------------------------------------------
__global__ void fps_kernel(const float* __restrict__ xyz,   // [B,N,3]
                           float* __restrict__ new_xyz,     // [B,S,3]
                           int N, int S) {
  int b = blockIdx.x;
  const float* p = xyz + (size_t)b * N * 3;
  __shared__ float dist[4096];
  __shared__ float redv[256];
  __shared__ int   redi[256];
  __shared__ int   s_far;
  int t = threadIdx.x;
  for (int i = t; i < N; i += blockDim.x) dist[i] = 1e10f;
  if (t == 0) s_far = 0;
  __syncthreads();
  for (int it = 0; it < S; ++it) {
    int far = s_far;
    if (t == 0) {
      new_xyz[((size_t)b * S + it) * 3 + 0] = p[far * 3 + 0];
      new_xyz[((size_t)b * S + it) * 3 + 1] = p[far * 3 + 1];
      new_xyz[((size_t)b * S + it) * 3 + 2] = p[far * 3 + 2];
    }
    float cx = p[far * 3 + 0], cy = p[far * 3 + 1], cz = p[far * 3 + 2];
    float best = -1.0f; int besti = 0x7fffffff;
    for (int i = t; i < N; i += blockDim.x) {
      float dx = p[i * 3 + 0] - cx;
      float dy = p[i * 3 + 1] - cy;
      float dz = p[i * 3 + 2] - cz;
      float d = dx * dx + dy * dy + dz * dz;
      float nd = fminf(dist[i], d);
      dist[i] = nd;
      if (nd > best) { best = nd; besti = i; }   // strict: keep earliest
    }
    redv[t] = best; redi[t] = besti;
    __syncthreads();
    for (int ofs = 128; ofs > 0; ofs >>= 1) {
      if (t < ofs) {
        float ov = redv[t + ofs]; int oi = redi[t + ofs];
        if (ov > redv[t] || (ov == redv[t] && oi < redi[t])) {
          redv[t] = ov; redi[t] = oi;
        }
      }
      __syncthreads();
    }
    if (t == 0) s_far = redi[0];
    __syncthreads();
  }
}

// ---------------------------------------------------------------------------
// Ball query: one thread per centroid; ascending scan == reference's
// where/sort/take-K; pad with first in-radius index.
// ---------------------------------------------------------------------------
__global__ void ball_query_kernel(const float* __restrict__ xyz,    // [B,N,3]
                                  const float* __restrict__ new_xyz,// [B,S,3]
                                  int* __restrict__ idx,            // [B*S*K]
                                  int B, int N, int S, int K, float r2) {
  int g = blockIdx.x * blockDim.x + threadIdx.x;
  if (g >= B * S) return;
  int b = g / S, s = g % S;
  const float* p = xyz + (size_t)b * N * 3;
  float cx = new_xyz[((size_t)b * S + s) * 3 + 0];
  float cy = new_xyz[((size_t)b * S + s) * 3 + 1];
  float cz = new_xyz[((size_t)b * S + s) * 3 + 2];
  int* o = idx + ((size_t)b * S + s) * K;
  int cnt = 0, first = 0;
  for (int i = 0; i < N && cnt < K; ++i) {
    float dx = p[i * 3 + 0] - cx;
    float dy = p[i * 3 + 1] - cy;
    float dz = p[i * 3 + 2] - cz;
    float d = dx * dx + dy * dy + dz * dz;
    if (d <= r2) { if (cnt == 0) first = i; o[cnt++] = i; }
  }
  for (; cnt < K; ++cnt) o[cnt] = first;
}

// ---------------------------------------------------------------------------
// Build grouped feature matrix X[C, Ntot], n = (b*S+s)*K + k.
// Channels: [0,Cp) gathered point features, [Cp,Cp+3) relative xyz.
// ---------------------------------------------------------------------------
__global__ void group_fill_kernel(const float* __restrict__ xyz,     // [B,Np,3]
                                  const float* __restrict__ new_xyz, // [B,S,3]
                                  const float* __restrict__ pts,     // [B,Cp,Np] or null
                                  const int* __restrict__ idx,       // [B*S*K]
                                  float* __restrict__ X,             // [C, Ntot]
                                  int B, int Np, int S, int K, int Cp) {
  int C = Cp + 3;
  long long Ntot = (long long)B * S * K;
  long long g = (long long)blockIdx.x * blockDim.x + threadIdx.x;
  if (g >= (long long)C * Ntot) return;
  long long n = g % Ntot;
  int c = (int)(g / Ntot);
  int j = idx[n];
  int sc = (int)((n / K) % S);
  int b  = (int)(n / ((long long)K * S));
  float v;
  if (c < Cp) {
    v = pts[((size_t)b * Cp + c) * Np + j];
  } else {
    int d = c - Cp;
    v = xyz[((size_t)b * Np + j) * 3 + d] - new_xyz[((size_t)b * S + sc) * 3 + d];
  }
  X[(size_t)c * Ntot + n] = v;
}

// ---------------------------------------------------------------------------
// WMMA f32 GEMM: Y[M,N] = W[M,C] * X[C,N].
// Each wave computes a 32(M) x 64(N) strip: 2 A-fragments x 4 B-fragments,
// 8 accumulators, 8 V_WMMA_F32_16X16X4_F32 per K-step.
// Main loop is guard-free (c0+4 <= C); single guarded tail handles C%4.
// Requires M % 32 == 0 and N % 64 == 0 (true for every layer here).
// A layout (16x4 f32): lane m=lane&15, VGPRs hold K = 2*(lane>>4)+{0,1}.
// B layout (4x16):     lane n=lane&15, same K halves.
// C/D: vgpr r -> row r + 8*(lane>>4), col lane&15.
// ---------------------------------------------------------------------------
__global__ void gemm_wmma_f32(const float* __restrict__ W,  // [M,C]
                              const float* __restrict__ X,  // [C,N]
                              float* __restrict__ Y,        // [M,N]
                              int M, long long Ntot, int C) {
  int lane = threadIdx.x & 31;
  int wave = threadIdx.x >> 5;
  long long tileId = (long long)blockIdx.x * (blockDim.x >> 5) + wave;
  int tilesM = M >> 5;               // 32-row strips
  long long tilesN = Ntot >> 6;      // 64-col strips
  if (tileId >= (long long)tilesM * tilesN) return;   // wave-uniform exit
  int tm = (int)(tileId % tilesM);
  long long tn = tileId / tilesM;

  int m0      = (tm << 5) + (lane & 15);
  int m1      = m0 + 16;
  long long n = (tn << 6) + (lane & 15);
  int kh      = (lane >> 4) << 1;                     // 0 or 2

  const float* w0 = W + (size_t)m0 * C;
  const float* w1 = W + (size_t)m1 * C;

  v8f acc[8] = {};                                    // [i*4+j]: i=M-tile, j=N-tile

  int c0 = 0;
  for (; c0 + 4 <= C; c0 += 4) {                      // guard-free main loop
    v2f a0, a1;
    a0.x = w0[c0 + kh];
    a0.y = w0[c0 + kh + 1];
    a1.x = w1[c0 + kh];
    a1.y = w1[c0 + kh + 1];
    const float* xa = X + (size_t)(c0 + kh) * Ntot + n;
    const float* xb = xa + Ntot;
#pragma unroll
    for (int j = 0; j < 4; ++j) {
      v2f bv;
      bv.x = xa[16 * j];
      bv.y = xb[16 * j];
      acc[j]     = __builtin_amdgcn_wmma_f32_16x16x4_f32(
          false, a0, false, bv, (short)0, acc[j],     false, false);
      acc[4 + j] = __builtin_amdgcn_wmma_f32_16x16x4_f32(
          false, a1, false, bv, (short)0, acc[4 + j], false, false);
    }
  }
  if (c0 < C) {                                       // tail: C % 4 != 0
    int ca = c0 + kh, cb = ca + 1;
    v2f a0 = {0.f, 0.f}, a1 = {0.f, 0.f};
    if (ca < C) { a0.x = w0[ca]; a1.x = w1[ca]; }
    if (cb < C) { a0.y = w0[cb]; a1.y = w1[cb]; }
#pragma unroll
    for (int j = 0; j < 4; ++j) {
      v2f bv = {0.f, 0.f};
      if (ca < C) bv.x = X[(size_t)ca * Ntot + n + 16 * j];
      if (cb < C) bv.y = X[(size_t)cb * Ntot + n + 16 * j];
      // EXEC re-converged here: WMMA runs with full mask as required.
      acc[j]     = __builtin_amdgcn_wmma_f32_16x16x4_f32(
          false, a0, false, bv, (short)0, acc[j],     false, false);
      acc[4 + j] = __builtin_amdgcn_wmma_f32_16x16x4_f32(
          false, a1, false, bv, (short)0, acc[4 + j], false, false);
    }
  }

  int rb = (tm << 5) + ((lane >> 4) << 3);
  long long colb = (tn << 6) + (lane & 15);
#pragma unroll
  for (int i = 0; i < 2; ++i) {
#pragma unroll
    for (int j = 0; j < 4; ++j) {
#pragma unroll
      for (int r = 0; r < 8; ++r) {
        Y[(size_t)(rb + 16 * i + r) * Ntot + (colb + 16 * j)] = acc[i * 4 + j][r];
      }
    }
  }
}

// ---------------------------------------------------------------------------
// Per-channel (row) mean + inv-std over Ntot columns. One block per row.
// ---------------------------------------------------------------------------
__global__ void bn_stats_kernel(const float* __restrict__ Y, float* __restrict__ stats,
                                long long Ntot) {
  int o = blockIdx.x;
  const float* row = Y + (size_t)o * Ntot;
  __shared__ float ss[256];
  __shared__ float sq[256];
  float s = 0.f, q = 0.f;
  for (long long i = threadIdx.x; i < Ntot; i += blockDim.x) {
    float v = row[i];
    s += v; q += v * v;
  }
  ss[threadIdx.x] = s; sq[threadIdx.x] = q;
  __syncthreads();
  for (int ofs = 128; ofs > 0; ofs >>= 1) {
    if ((int)threadIdx.x < ofs) {
      ss[threadIdx.x] += ss[threadIdx.x + ofs];
      sq[threadIdx.x] += sq[threadIdx.x + ofs];
    }
    __syncthreads();
  }
  if (threadIdx.x == 0) {
    float m   = ss[0] / (float)Ntot;
    float var = sq[0] / (float)Ntot - m * m;
    stats[2 * o]     = m;
    stats[2 * o + 1] = rsqrtf(var + BN_EPS);
  }
}

// 2-D grid: blockIdx.y = channel, blockIdx.x covers columns (no div/mod).
__global__ void bn_relu_kernel(float* __restrict__ Y, const float* __restrict__ stats,
                               long long Ntot) {
  int o = blockIdx.y;
  long long i = (long long)blockIdx.x * blockDim.x + threadIdx.x;
  if (i >= Ntot) return;
  float m = stats[2 * o], is = stats[2 * o + 1];
  float* row = Y + (size_t)o * Ntot;
  float v = (row[i] - m) * is;
  row[i] = v > 0.f ? v : 0.f;
}

// ---------------------------------------------------------------------------
// Max over K (contiguous runs since n=(b*S+s)*K+k).
// out[(b*Ctot + coff + o)*S + s] = max_k Y[o][(b*S+s)*K+k]
// ---------------------------------------------------------------------------
__global__ void maxpool_kernel(const float* __restrict__ Y, float* __restrict__ out,
                               int B, int S, int K, int O, int Ctot, int coff) {
  long long Ntot = (long long)B * S * K;
  long long g = (long long)blockIdx.x * blockDim.x + threadIdx.x;
  long long total = (long long)B * O * S;
  if (g >= total) return;
  int s = (int)(g % S);
  int o = (int)((g / S) % O);
  int b = (int)(g / ((long long)S * O));
  const float* row = Y + (size_t)o * Ntot + ((size_t)b * S + s) * K;
  float m = row[0];
  for (int k = 1; k < K; ++k) m = fmaxf(m, row[k]);
  out[((size_t)b * Ctot + coff + o) * S + s] = m;
}

// ---------------------------------------------------------------------------
// sa_group_all input: X[643, B*128], channels = [xyz(3), points(640)].
// n = b*128 + s.
// ---------------------------------------------------------------------------
__global__ void sa3_fill_kernel(const float* __restrict__ l2_xyz,  // [B,128,3]
                                const float* __restrict__ l2_pts,  // [B,640,128]
                                float* __restrict__ X) {            // [643,1024]
  const int Ntot = 8 * 128;
  int g = blockIdx.x * blockDim.x + threadIdx.x;
  if (g >= 643 * Ntot) return;
  int n = g % Ntot;
  int c = g / Ntot;
  int b = n / 128, s = n % 128;
  float v = (c < 3) ? l2_xyz[((size_t)b * 128 + s) * 3 + c]
                    : l2_pts[((size_t)b * 640 + (c - 3)) * 128 + s];
  X[(size_t)c * Ntot + n] = v;
}

// ---------------------------------------------------------------------------
// Small FC: y[b,o] = sum_i h[b,i]*W[o,i] + bias[o]
// ---------------------------------------------------------------------------
__global__ void fc_kernel(const float* __restrict__ h, const float* __restrict__ W,
                          const float* __restrict__ bias, float* __restrict__ y,
                          int Bn, int In, int Out) {
  int g = blockIdx.x * blockDim.x + threadIdx.x;
  if (g >= Bn * Out) return;
  int b = g / Out, o = g % Out;
  float s = bias ? bias[o] : 0.0f;
  const float* hr = h + (size_t)b * In;
  const float* wr = W + (size_t)o * In;
  for (int i = 0; i < In; ++i) s += hr[i] * wr[i];
  y[g] = s;
}

// BN over batch axis (8 rows) per feature column, then ReLU, in place.
__global__ void bn_relu_cols_kernel(float* __restrict__ y, int Bn, int Out) {
  int o = blockIdx.x * blockDim.x + threadIdx.x;
  if (o >= Out) return;
  float s = 0.f, q = 0.f;
  for (int b = 0; b < Bn; ++b) { float v = y[(size_t)b * Out + o]; s += v; q += v * v; }
  float m  = s / (float)Bn;
  float is = rsqrtf(q / (float)Bn - m * m + BN_EPS);
  for (int b = 0; b < Bn; ++b) {
    float v = (y[(size_t)b * Out + o] - m) * is;
    y[(size_t)b * Out + o] = v > 0.f ? v : 0.f;
  }
}

// ---------------------------------------------------------------------------
// Host orchestration.
// d_in layout (dict order, tuples flattened):
//   0: xyz [8,4096,3]
//   1..18 : SA1  (3 branches x 3 layers x (W,b))   base = 1  + 6*branch
//   19..36: SA2                                    base = 19 + 6*branch
//   37..42: SA3  (3 layers x (W,b))                base = 37
//   43..48: FC   (3 x (W,b))                       base = 43
// Conv biases are zeros AND cancel under the following batch-norm, so conv
// layers are pure GEMM + BN + ReLU.
// ---------------------------------------------------------------------------
extern "C" void kernel_launch(void* const* d_in, const int* in_sizes, int n_in,
                              void* d_out, int out_size, void* d_ws, size_t ws_size,
                              hipStream_t stream) {
  (void)in_sizes; (void)n_in; (void)out_size; (void)ws_size;
  const float* xyz = (const float*)d_in[0];
  const int B = 8, N0 = 4096;

  // ---- workspace carving ----
  char* wp = (char*)d_ws;
  auto carve = [&](size_t bytes) -> void* {
    void* p = (void*)wp;
    wp += (bytes + 255) & ~(size_t)255;
    return p;
  };
  float* l1_xyz = (float*)carve((size_t)B * 512 * 3 * 4);
  float* l2_xyz = (float*)carve((size_t)B * 128 * 3 * 4);
  int*   idxb   = (int*)  carve((size_t)B * 512 * 128 * 4);
  float* stats  = (float*)carve((size_t)2 * 1024 * 4);
  float* l1_pts = (float*)carve((size_t)B * 320 * 512 * 4);
  float* l2_pts = (float*)carve((size_t)B * 640 * 128 * 4);
  float* h0     = (float*)carve((size_t)B * 1024 * 4);
  float* h1     = (float*)carve((size_t)B * 512 * 4);
  float* h2     = (float*)carve((size_t)B * 256 * 4);
  const size_t bigElems = (size_t)128 * 524288;   // largest intermediate
  float* bufA = (float*)carve(bigElems * 4);
  float* bufB = (float*)carve(bigElems * 4);

  // ---- conv MLP runner (GEMM + BN + ReLU, ping-pong) ----
  auto run_mlp = [&](float* Xin, float* Xalt, int cin, const int* widths,
                     int pbase, long long Ntot) -> float* {
    float* in = Xin; float* out = Xalt;
    for (int l = 0; l < 3; ++l) {
      int M = widths[l];
      const float* W = (const float*)d_in[pbase + 2 * l];
      long long tiles = (long long)(M / 32) * (Ntot / 64);
      unsigned gblocks = (unsigned)((tiles + 7) / 8);      // 8 waves/block
      gemm_wmma_f32<<<gblocks, 256, 0, stream>>>(W, in, out, M, Ntot, cin);
      bn_stats_kernel<<<M, 256, 0, stream>>>(out, stats, Ntot);
      dim3 bgrid((unsigned)((Ntot + 255) / 256), (unsigned)M);
      bn_relu_kernel<<<bgrid, 256, 0, stream>>>(out, stats, Ntot);
      float* t = in; in = out; out = t;
      cin = M;
    }
    return in;   // final activation after 3 swaps
  };

  // ================= SA1 (points = xyz only, Cp = 0) =================
  fps_kernel<<<B, 256, 0, stream>>>(xyz, l1_xyz, N0, 512);
  {
    const int   Ks[3]  = {16, 32, 128};
    const float r2s[3] = {0.01f, 0.04f, 0.16f};
    const int w0[3] = {32, 32, 64}, w1[3] = {64, 64, 128}, w2[3] = {64, 96, 128};
    const int* wlist[3] = {w0, w1, w2};
    const int coffs[3] = {0, 64, 192};   // concat into 320 channels
    for (int br = 0; br < 3; ++br) {
      int K = Ks[br]; int S = 512;
      ball_query_kernel<<<(B * S + 255) / 256, 256, 0, stream>>>(
          xyz, l1_xyz, idxb, B, N0, S, K, r2s[br]);
      long long Ntot = (long long)B * S * K;
      long long fill = 3 * Ntot;
      group_fill_kernel<<<(unsigned)((fill + 255) / 256), 256, 0, stream>>>(
          xyz, l1_xyz, nullptr, idxb, bufA, B, N0, S, K, /*Cp=*/0);
      float* fin = run_mlp(bufA, bufB, 3, wlist[br], 1 + 6 * br, Ntot);
      int O = wlist[br][2];
      long long mt = (long long)B * O * S;
      maxpool_kernel<<<(unsigned)((mt + 255) / 256), 256, 0, stream>>>(
          fin, l1_pts, B, S, K, O, 320, coffs[br]);
    }
  }

  // ================= SA2 (points = l1_pts, Cp = 320) =================
  fps_kernel<<<B, 256, 0, stream>>>(l1_xyz, l2_xyz, 512, 128);
  {
    const int   Ks[3]  = {32, 64, 128};
    const float r2s[3] = {0.04f, 0.16f, 0.64f};
    const int w0[3] = {64, 64, 128}, w1[3] = {128, 128, 256}, w2[3] = {128, 128, 256};
    const int* wlist[3] = {w0, w1, w2};
    const int coffs[3] = {0, 128, 384};  // concat into 640 channels
    for (int br = 0; br < 3; ++br) {
      int K = Ks[br]; int S = 128;
      ball_query_kernel<<<(B * S + 255) / 256, 256, 0, stream>>>(
          l1_xyz, l2_xyz, idxb, B, 512, S, K, r2s[br]);
      long long Ntot = (long long)B * S * K;
      long long fill = (long long)(320 + 3) * Ntot;
      group_fill_kernel<<<(unsigned)((fill + 255) / 256), 256, 0, stream>>>(
          l1_xyz, l2_xyz, l1_pts, idxb, bufA, B, 512, S, K, /*Cp=*/320);
      float* fin = run_mlp(bufA, bufB, 323, wlist[br], 19 + 6 * br, Ntot);
      int O = wlist[br][2];
      long long mt = (long long)B * O * S;
      maxpool_kernel<<<(unsigned)((mt + 255) / 256), 256, 0, stream>>>(
          fin, l2_pts, B, S, K, O, 640, coffs[br]);
    }
  }

  // ================= SA3 (group all) =================
  {
    sa3_fill_kernel<<<(643 * 1024 + 255) / 256, 256, 0, stream>>>(l2_xyz, l2_pts, bufA);
    const int w3[3] = {256, 512, 1024};
    float* fin = run_mlp(bufA, bufB, 643, w3, 37, /*Ntot=*/1024);
    // max over the 128 samples per batch: S=1, K=128 -> feat [8,1024]
    maxpool_kernel<<<(8 * 1024 + 255) / 256, 256, 0, stream>>>(
        fin, h0, B, /*S=*/1, /*K=*/128, /*O=*/1024, /*Ctot=*/1024, /*coff=*/0);
  }

  // ================= FC head =================
  fc_kernel<<<(8 * 512 + 255) / 256, 256, 0, stream>>>(
      h0, (const float*)d_in[43], (const float*)d_in[44], h1, 8, 1024, 512);
  bn_relu_cols_kernel<<<(512 + 255) / 256, 256, 0, stream>>>(h1, 8, 512);

  fc_kernel<<<(8 * 256 + 255) / 256, 256, 0, stream>>>(
      h1, (const float*)d_in[45], (const float*)d_in[46], h2, 8, 512, 256);
  bn_relu_cols_kernel<<<(256 + 255) / 256, 256, 0, stream>>>(h2, 8, 256);

  fc_kernel<<<(8 * 256 + 255) / 256, 256, 0, stream>>>(
      h2, (const float*)d_in[47], (const float*)d_in[48], (float*)d_out, 8, 256, 256);
}